// HistogramBinningCalibrationByFeature_6468220747979
// MI455X (gfx1250) — compile-verified
//
#include <hip/hip_runtime.h>
#include <hip/hip_bf16.h>
#include <stdint.h>

#define RECAL_C 0.9162907600402832f

__device__ __forceinline__ float calib_one(
    float x, int sv, int sl, int nseg, int nbins, float fb,
    const float* __restrict__ sh, int nb,
    const float* __restrict__ bin_pos, const float* __restrict__ bin_ex)
{
    // recalibrated sigmoid
    float o = 1.0f / (1.0f + expf(RECAL_C - x));

    // dense segment id (+1), clamped; zero when length != 1
    int s = sv + 1;
    if (s > nseg || s < 0) s = 0;
    if (sl != 1) s = 0;

    // searchsorted(boundaries, o, side='left') over a uniform grid:
    // analytic guess + exact fixup against the LDS copy (0-1 iters each way)
    int g = (int)(o * fb);
    if (g < 0) g = 0;
    if (g > nb) g = nb;
    while (g > 0 && sh[g - 1] >= o) --g;
    while (g < nb && sh[g] < o) ++g;

    long idx = (long)g + (long)s * (long)nbins;
    float p = bin_pos[idx];
    float e = bin_ex[idx];
    float cal = (p / e) * 0.9995f + o * 0.0005f;
    return (e > 10000.0f) ? cal : o;
}

__global__ __launch_bounds__(256) void hbc_by_feature_kernel(
    const float* __restrict__ logit,
    const float* __restrict__ bin_pos,
    const float* __restrict__ bin_ex,
    const float* __restrict__ bnd,
    const int*   __restrict__ segv,
    const int*   __restrict__ segl,
    const int*   __restrict__ p_nseg,
    const int*   __restrict__ p_nbins,
    float* __restrict__ out,
    int n, int nb)
{
    extern __shared__ float sh[];
    const int tid = threadIdx.x;

    // --- Stage boundary table (~20KB) into LDS via CDNA5 async DMA ---
    // GLOBAL_LOAD_ASYNC_TO_LDS_B32 (GVS mode: SGPR64 base + VGPR32 offset),
    // tracked by ASYNCcnt; drained with s_wait_asynccnt, then barrier.
    for (int i = tid; i < nb; i += (int)blockDim.x) {
        unsigned lds_addr = (unsigned)(uintptr_t)(&sh[i]); // low 32 bits of flat shared addr == LDS byte addr
        unsigned goff     = (unsigned)(i * 4);
        asm volatile("global_load_async_to_lds_b32 %0, %1, %2"
                     :
                     : "v"(lds_addr), "v"(goff), "s"(bnd)
                     : "memory");
    }
    asm volatile("s_wait_asynccnt 0" ::: "memory");
    __syncthreads();

    const int nseg  = *p_nseg;   // uniform -> scalar load
    const int nbins = *p_nbins;
    const float fb  = (float)nbins;

    int base = ((int)blockIdx.x * (int)blockDim.x + tid) * 4;
    if (base >= n) return;

    if (base + 3 < n) {
        // vectorized b128 path
        float4 x4  = *(const float4*)(logit + base);
        int4   sv4 = *(const int4*)(segv + base);
        int4   sl4 = *(const int4*)(segl + base);
        float4 r;
        r.x = calib_one(x4.x, sv4.x, sl4.x, nseg, nbins, fb, sh, nb, bin_pos, bin_ex);
        r.y = calib_one(x4.y, sv4.y, sl4.y, nseg, nbins, fb, sh, nb, bin_pos, bin_ex);
        r.z = calib_one(x4.z, sv4.z, sl4.z, nseg, nbins, fb, sh, nb, bin_pos, bin_ex);
        r.w = calib_one(x4.w, sv4.w, sl4.w, nseg, nbins, fb, sh, nb, bin_pos, bin_ex);
        *(float4*)(out + base) = r;
    } else {
        for (int i = base; i < n && i < base + 4; ++i) {
            out[i] = calib_one(logit[i], segv[i], segl[i], nseg, nbins, fb, sh, nb,
                               bin_pos, bin_ex);
        }
    }
}

extern "C" void kernel_launch(void* const* d_in, const int* in_sizes, int n_in,
                              void* d_out, int out_size, void* d_ws, size_t ws_size,
                              hipStream_t stream) {
    const float* logit = (const float*)d_in[0];
    const float* pos   = (const float*)d_in[1];
    const float* ex    = (const float*)d_in[2];
    const float* bnd   = (const float*)d_in[3];
    const int*   segv  = (const int*)d_in[4];
    const int*   segl  = (const int*)d_in[5];
    const int*   pns   = (const int*)d_in[6];   // num_segments (scalar)
    const int*   pnb   = (const int*)d_in[7];   // num_bins (scalar)
    float* out = (float*)d_out;

    int n  = in_sizes[0];   // N (logit is N x 1)
    int nb = in_sizes[3];   // number of boundaries (num_bins - 1)

    int threads = 256;
    int elems_per_block = threads * 4;
    int blocks = (n + elems_per_block - 1) / elems_per_block;
    size_t shmem = (size_t)nb * sizeof(float);

    hbc_by_feature_kernel<<<blocks, threads, shmem, stream>>>(
        logit, pos, ex, bnd, segv, segl, pns, pnb, out, n, nb);
}